// GCN_65231963292247
// MI455X (gfx1250) — compile-verified
//
#include <hip/hip_runtime.h>

typedef float v2f __attribute__((ext_vector_type(2)));
typedef float v8f __attribute__((ext_vector_type(8)));

#define F_IN  128
#define F_HID 100
#define LDS_K    128
#define LDS_COLS 112            // 7 column tiles of 16, cols 100..111 zero-padded
#define MT_PER_BLK 8            // 8 waves = 8 row tiles per workgroup

// ---------- degree / normalization ----------
__global__ __launch_bounds__(256) void k_init_deg(float* __restrict__ deg, int n) {
  int i = blockIdx.x * 256 + threadIdx.x;
  if (i < n) deg[i] = 1.0f;                     // self-loop contribution
}

__global__ __launch_bounds__(256) void k_deg_edges(const int* __restrict__ dst,
                                                   float* __restrict__ deg, int e) {
  int i = blockIdx.x * 256 + threadIdx.x;
  if (i < e) atomicAdd(&deg[dst[i]], 1.0f);
}

__global__ __launch_bounds__(256) void k_dinv(float* __restrict__ deg, int n) {
  int i = blockIdx.x * 256 + threadIdx.x;
  if (i < n) deg[i] = rsqrtf(deg[i]);           // deg >= 1 always (self-loops)
}

// ---------- layer 1 GEMM: h[N,100] = x[N,128] @ W1[128,100] via f32 WMMA ----------
// W1 staged transposed + zero-padded in LDS: Wt[col][k] so a lane's (K,K+1) B pair
// is one 8B-aligned ds_load_b64 and the pad columns make the loop guard-free.
// Each wave owns one 16-row tile; A slice preloaded into 32 v2f registers and
// reused across the 7 column tiles. Inner loop = ds_load_b64 + v_wmma only.
__global__ __launch_bounds__(256) void k_gemm1(const float* __restrict__ x,
                                               const float* __restrict__ W1,
                                               float* __restrict__ h, int mtiles) {
  __shared__ float Wt[LDS_COLS * LDS_K];        // 57,344 B of 320 KB

  const int tid = threadIdx.x;
  // coalesced read of W1 (k-major), transposed scatter into LDS
  for (int idx = tid; idx < F_IN * F_HID; idx += 256) {
    const int k = idx / F_HID;
    const int c = idx - k * F_HID;              // idx = k*100 + c
    Wt[c * LDS_K + k] = W1[idx];
  }
  // zero the pad columns 100..111
  for (int idx = tid; idx < (LDS_COLS - F_HID) * LDS_K; idx += 256) {
    const int c = F_HID + (idx >> 7);
    const int k = idx & (LDS_K - 1);
    Wt[c * LDS_K + k] = 0.0f;
  }
  __syncthreads();

  const int wave = tid >> 5;
  const int lane = tid & 31;
  const int r    = lane & 15;
  const int khi  = lane >> 4;                   // 0 or 1
  const int mt   = blockIdx.x * MT_PER_BLK + wave;
  if (mt >= mtiles) return;
  const int row = mt * 16 + r;

  // preload this lane's A slice: K pairs {4s+2khi, 4s+2khi+1}, s = 0..31
  const float* xrow = x + (size_t)row * F_IN + 2 * khi;
  v2f a[32];
#pragma unroll
  for (int s = 0; s < 32; ++s) a[s] = *(const v2f*)(xrow + 4 * s);

  for (int ct = 0; ct < 7; ++ct) {
    const int col = ct * 16 + r;                // B/C: lane -> matrix col
    const float* bcol = &Wt[col * LDS_K + 2 * khi];
    v8f acc = {};
#pragma unroll
    for (int s = 0; s < 32; ++s) {
      const v2f b = *(const v2f*)(bcol + 4 * s);
      acc = __builtin_amdgcn_wmma_f32_16x16x4_f32(false, a[s], false, b,
                                                  (short)0, acc, false, false);
    }
    if (col < F_HID) {                          // only store guard remains
      float* hp = h + ((size_t)mt * 16 + 8 * khi) * F_HID + col;
#pragma unroll
      for (int i = 0; i < 8; ++i) hp[(size_t)i * F_HID] = acc[i];
    }
  }
}

// ---------- propagation 1: agg = b1 + self-loop + scatter-add over edges ----------
__global__ __launch_bounds__(128) void k_self1(const float* __restrict__ h,
                                               const float* __restrict__ dinv,
                                               const float* __restrict__ b1,
                                               float* __restrict__ agg) {
  const int i = blockIdx.x;
  const int f = threadIdx.x;
  if (f < F_HID) {
    const float d = dinv[i];
    agg[(size_t)i * F_HID + f] = b1[f] + h[(size_t)i * F_HID + f] * d * d;
  }
}

// One block per edge: src/dst/norm are wave-uniform -> scalar loads; 100 lanes
// do a coalesced gather of h[src,:] and global_atomic_add_f32 into agg[dst,:].
__global__ __launch_bounds__(128) void k_edge1(const int* __restrict__ src,
                                               const int* __restrict__ dst,
                                               const float* __restrict__ dinv,
                                               const float* __restrict__ h,
                                               float* __restrict__ agg) {
  const int e = blockIdx.x;
  const int s = src[e];
  const int d = dst[e];
  const float norm = dinv[s] * dinv[d];
  const int f = threadIdx.x;
  if (f < F_HID)
    atomicAdd(&agg[(size_t)d * F_HID + f], h[(size_t)s * F_HID + f] * norm);
}

// ---------- layer 2: h2[i] = relu(agg[i,:]) . W2  (one wave32 per node) ----------
__global__ __launch_bounds__(256) void k_matvec(const float* __restrict__ agg,
                                                const float* __restrict__ W2,
                                                float* __restrict__ h2, int n) {
  const int gid  = blockIdx.x * 256 + threadIdx.x;
  const int node = gid >> 5;
  const int lane = gid & 31;
  if (node >= n) return;
  const float* row = agg + (size_t)node * F_HID;
  float sum = 0.0f;
#pragma unroll
  for (int f0 = 0; f0 < F_HID; f0 += 32) {
    const int f = f0 + lane;
    if (f < F_HID) {
      float v = fmaxf(row[f], 0.0f);            // relu
      sum += v * W2[f];
    }
  }
#pragma unroll
  for (int off = 16; off; off >>= 1) sum += __shfl_xor(sum, off, 32);
  if (lane == 0) h2[node] = sum;
}

// ---------- propagation 2 (scalar features) ----------
__global__ __launch_bounds__(256) void k_self2(const float* __restrict__ h2,
                                               const float* __restrict__ dinv,
                                               const float* __restrict__ b2,
                                               float* __restrict__ out, int n) {
  int i = blockIdx.x * 256 + threadIdx.x;
  if (i < n) {
    const float d = dinv[i];
    out[i] = b2[0] + h2[i] * d * d;
  }
}

__global__ __launch_bounds__(256) void k_edge2(const int* __restrict__ src,
                                               const int* __restrict__ dst,
                                               const float* __restrict__ dinv,
                                               const float* __restrict__ h2,
                                               float* __restrict__ out, int e) {
  int i = blockIdx.x * 256 + threadIdx.x;
  if (i < e) {
    const int s = src[i];
    const int d = dst[i];
    atomicAdd(&out[d], h2[s] * dinv[s] * dinv[d]);
  }
}

extern "C" void kernel_launch(void* const* d_in, const int* in_sizes, int n_in,
                              void* d_out, int out_size, void* d_ws, size_t ws_size,
                              hipStream_t stream) {
  (void)n_in; (void)out_size; (void)ws_size;
  const float* x  = (const float*)d_in[0];
  const int*   ei = (const int*)d_in[1];       // [2, E] (int32 — JAX default x32)
  const float* W1 = (const float*)d_in[2];
  const float* b1 = (const float*)d_in[3];
  const float* W2 = (const float*)d_in[4];
  const float* b2 = (const float*)d_in[5];
  float* out = (float*)d_out;

  const int N = in_sizes[0] / F_IN;            // 100000
  const int E = in_sizes[1] / 2;               // 3200000
  const int* src = ei;
  const int* dst = ei + E;

  // workspace layout (256B-aligned slices)
  char* ws = (char*)d_ws;
  size_t o = 0;
  float* dinv = (float*)(ws + o); o += (((size_t)N * 4)          + 255) & ~(size_t)255;
  float* h    = (float*)(ws + o); o += (((size_t)N * F_HID * 4)  + 255) & ~(size_t)255;
  float* agg  = (float*)(ws + o); o += (((size_t)N * F_HID * 4)  + 255) & ~(size_t)255;
  float* h2   = (float*)(ws + o);

  const int nb = (N + 255) / 256;
  const int eb = (E + 255) / 256;

  k_init_deg <<<nb, 256, 0, stream>>>(dinv, N);
  k_deg_edges<<<eb, 256, 0, stream>>>(dst, dinv, E);
  k_dinv     <<<nb, 256, 0, stream>>>(dinv, N);

  const int mtiles = N / 16;                   // 6250
  const int gblocks = (mtiles + MT_PER_BLK - 1) / MT_PER_BLK;  // 782
  k_gemm1    <<<gblocks, 256, 0, stream>>>(x, W1, h, mtiles);

  k_self1    <<<N, 128, 0, stream>>>(h, dinv, b1, agg);
  k_edge1    <<<E, 128, 0, stream>>>(src, dst, dinv, h, agg);

  k_matvec   <<<(N * 32 + 255) / 256, 256, 0, stream>>>(agg, W2, h2, N);

  k_self2    <<<nb, 256, 0, stream>>>(h2, dinv, b2, out, N);
  k_edge2    <<<eb, 256, 0, stream>>>(src, dst, dinv, h2, out, E);
}